// SSIM_47485158424861
// MI455X (gfx1250) — compile-verified
//
#include <hip/hip_runtime.h>

typedef __attribute__((ext_vector_type(2))) float v2f;
typedef __attribute__((ext_vector_type(8))) float v8f;

#define TILE   32
#define HALO   5
#define IN_T   42          // TILE + 2*HALO
#define XS     44          // x/y tile row stride (floats)
#define HS     40          // h-field row stride: 2*HS*4B -> +16 banks between WMMA B half-waves
#define RS     36          // result row stride:  8*RS*4B -> +32 banks between WMMA C half-waves
#define IMG_H  512
#define IMG_W  512

// 11-tap Gaussian, sigma=1.5, normalized (matches kornia get_gaussian_kernel1d)
#define G0 0.00102838f
#define G1 0.00759870f
#define G2 0.03600077f
#define G3 0.10936068f
#define G4 0.21300553f
#define G5 0.26601172f

__global__ __launch_bounds__(256) void ssim_fused_kernel(const float* __restrict__ img1,
                                                         const float* __restrict__ img2,
                                                         float* __restrict__ out,
                                                         int planes) {
    // LDS carve: hbuf | gpad | (xs,ys  overlapped with  rbuf)
    // hbuf: 5*42*40 = 8400, gpad: 64, region: max(2*42*44, 5*32*36) = 5760
    __shared__ __align__(16) float smem[8400 + 64 + 5760];   // 56,896 bytes
    float* hbuf = smem;                     // 5 fields x [42][HS]
    float* gpad = smem + 5 * IN_T * HS;     // 64-entry zero-padded gaussian
    float* xs   = gpad + 64;                // [42][XS]
    float* ys   = xs + IN_T * XS;
    float* rbuf = gpad + 64;                // 5 fields x [32][RS]  (aliases xs/ys)

    const int tid  = threadIdx.x;
    const int lane = tid & 31;
    const int M    = lane & 15;             // row-in-subtile / column index for WMMA operands
    const int hi   = lane >> 4;             // half-wave select

    const int c0    = blockIdx.x * TILE;
    const int r0    = blockIdx.y * TILE;
    const int plane = blockIdx.z;
    const size_t pbase = (size_t)plane * (IMG_H * IMG_W);
    const float* p1 = img1 + pbase;
    const float* p2 = img2 + pbase;

    // ---- Phase 0: zero-padded gaussian table gpad[16+t] = g[t] ----
    if (tid < 64) {
        int t = tid - 16;
        float v = 0.0f;
        if (t >= 0 && t <= 10) {
            int d = t - 5; if (d < 0) d = -d;
            v = (d == 0) ? G5 : (d == 1) ? G4 : (d == 2) ? G3
              : (d == 3) ? G2 : (d == 4) ? G1 : G0;
        }
        gpad[tid] = v;
    }

    // ---- Phase 1: cooperative halo-tile load with reflect padding ----
    for (int i = tid; i < IN_T * IN_T; i += 256) {
        int r = i / IN_T;
        int c = i - r * IN_T;
        int gr = r0 - HALO + r;
        gr = (gr < 0) ? -gr : gr;
        gr = (gr >= IMG_H) ? (2 * IMG_H - 2 - gr) : gr;
        int gc = c0 - HALO + c;
        gc = (gc < 0) ? -gc : gc;
        gc = (gc >= IMG_W) ? (2 * IMG_W - 2 - gc) : gc;
        float a = p1[gr * IMG_W + gc];
        float b = p2[gr * IMG_W + gc];
        xs[r * XS + c] = a;
        ys[r * XS + c] = b;
    }
    __syncthreads();

    // ---- Phase 2: horizontal 11-tap gaussian on 5 fields (VALU pipe) ----
    {
        const float GW[11] = {G0, G1, G2, G3, G4, G5, G4, G3, G2, G1, G0};
        for (int i = tid; i < IN_T * TILE; i += 256) {
            int r = i >> 5;
            int c = i & 31;
            const float* xr = &xs[r * XS + c];
            const float* yr = &ys[r * XS + c];
            float sx = 0.f, sy = 0.f, sxx = 0.f, syy = 0.f, sxy = 0.f;
#pragma unroll
            for (int t = 0; t < 11; ++t) {
                float w  = GW[t];
                float xv = xr[t];
                float yv = yr[t];
                sx  += w * xv;
                sy  += w * yv;
                sxx += w * xv * xv;
                syy += w * yv * yv;
                sxy += w * xv * yv;
            }
            int o = r * HS + c;
            hbuf[0 * IN_T * HS + o] = sx;
            hbuf[1 * IN_T * HS + o] = sy;
            hbuf[2 * IN_T * HS + o] = sxx;
            hbuf[3 * IN_T * HS + o] = syy;
            hbuf[4 * IN_T * HS + o] = sxy;
        }
    }
    __syncthreads();

    // ---- Phase 3: vertical gaussian as banded matmul on the matrix pipe ----
    // out[16x16] = A[16x28] x h[28x16], A[i][j] = g[j-i] (band), K padded 26->28.
    // A layout (32-bit A 16x4): lane L: M = L%16; VGPR0 holds K = 4b + 2*hi, VGPR1 holds K+1.
    float ax[7], ay[7];
#pragma unroll
    for (int b = 0; b < 7; ++b) {
        int j0 = 4 * b + 2 * hi;
        ax[b] = gpad[j0 - M + 16];
        ay[b] = gpad[j0 + 1 - M + 16];
    }

    int wave = __builtin_amdgcn_readfirstlane(tid >> 5);
    for (int j = wave; j < 20; j += 8) {           // 20 jobs = 5 fields x 4 subtiles
        int f  = j >> 2;
        int s  = j & 3;
        int sr = (s >> 1) << 4;                    // subtile row offset in output tile
        int sc = (s & 1) << 4;                     // subtile col offset
        const float* hf = hbuf + f * IN_T * HS;

        v8f acc = {0.f, 0.f, 0.f, 0.f, 0.f, 0.f, 0.f, 0.f};
#pragma unroll
        for (int b = 0; b < 7; ++b) {
            int j0   = 4 * b + 2 * hi;
            int row0 = sr + j0;     if (row0 > IN_T - 1) row0 = IN_T - 1;  // A is 0 there
            int row1 = sr + j0 + 1; if (row1 > IN_T - 1) row1 = IN_T - 1;
            v2f av = {ax[b], ay[b]};
            v2f bv;
            bv.x = hf[row0 * HS + sc + M];         // B: row K striped across lanes (N = lane%16)
            bv.y = hf[row1 * HS + sc + M];
            acc = __builtin_amdgcn_wmma_f32_16x16x4_f32(
                /*neg_a=*/false, av, /*neg_b=*/false, bv,
                /*c_mod=*/(short)0, acc, /*reuse_a=*/false, /*reuse_b=*/false);
        }
        // C/D layout: VGPR v -> row M=v (lanes 0-15), row M=v+8 (lanes 16-31), N = lane%16
        float* rf = rbuf + f * TILE * RS;
#pragma unroll
        for (int v = 0; v < 8; ++v)
            rf[(sr + hi * 8 + v) * RS + sc + M] = acc[v];
    }
    __syncthreads();

    // ---- Phase 4: elementwise SSIM and coalesced store ----
    const float C1v = 0.0001f;                      // (0.01*1.0)^2
    const float C2v = 0.0009f;                      // (0.03*1.0)^2
    const float EPSv = 1e-12f;
    float* po = out + pbase;
#pragma unroll
    for (int k = 0; k < 4; ++k) {
        int p   = tid + (k << 8);
        int row = p >> 5;
        int col = p & 31;
        int o   = row * RS + col;
        float mu1 = rbuf[0 * TILE * RS + o];
        float mu2 = rbuf[1 * TILE * RS + o];
        float exx = rbuf[2 * TILE * RS + o];
        float eyy = rbuf[3 * TILE * RS + o];
        float exy = rbuf[4 * TILE * RS + o];
        float mu1s = mu1 * mu1;
        float mu2s = mu2 * mu2;
        float mu12 = mu1 * mu2;
        float s1  = exx - mu1s;
        float s2  = eyy - mu2s;
        float s12 = exy - mu12;
        float num = (2.0f * mu12 + C1v) * (2.0f * s12 + C2v);
        float den = (mu1s + mu2s + C1v) * (s1 + s2 + C2v);
        po[(size_t)(r0 + row) * IMG_W + (c0 + col)] = num / (den + EPSv);
    }
}

extern "C" void kernel_launch(void* const* d_in, const int* in_sizes, int n_in,
                              void* d_out, int out_size, void* d_ws, size_t ws_size,
                              hipStream_t stream) {
    (void)n_in; (void)out_size; (void)d_ws; (void)ws_size;
    const float* img1 = (const float*)d_in[0];
    const float* img2 = (const float*)d_in[1];
    float* out = (float*)d_out;
    int planes = in_sizes[0] / (IMG_H * IMG_W);     // 16*3 = 48 for the reference shape
    dim3 grid(IMG_W / TILE, IMG_H / TILE, planes);  // 16 x 16 x 48 = 12288 workgroups
    ssim_fused_kernel<<<grid, 256, 0, stream>>>(img1, img2, out, planes);
}